// DMamba_49959059587624
// MI455X (gfx1250) — compile-verified
//
#include <hip/hip_runtime.h>
#include <math.h>

typedef __attribute__((ext_vector_type(2))) float v2f;
typedef __attribute__((ext_vector_type(8))) float v8f;

#define B_DIM   2
#define L_DIM   1024
#define DMODEL  1024
#define DINNER  2048
#define DSTATE  16
#define DTRANK  64
#define DCONV   4
#define M_ROWS  (B_DIM * L_DIM)   // 2048 token rows

__device__ __forceinline__ v8f wmma_f32(v2f a, v2f b, v8f c) {
    // D = A(16x4 f32) x B(4x16 f32) + C(16x16 f32)
    return __builtin_amdgcn_wmma_f32_16x16x4_f32(
        /*neg_a=*/false, a, /*neg_b=*/false, b,
        /*c_mod=*/(short)0, c, /*reuse_a=*/false, /*reuse_b=*/false);
}

// C[m][n] = sum_k A[m][k] * Bw[n][k]   (A row-major MxK, Bw row-major NxK)
// One wave per block; each wave computes a (16*MU) x (16*NU) tile:
// MU*NU accumulators, MU+NU fragment loads per K-step of 4.
// K must be a multiple of 16.
// EPI 0: plain store.  EPI 1: softplus(v + bias[n]).
template <int MU, int NU, int EPI>
__global__ void gemm_nt_kernel(const float* __restrict__ A,
                               const float* __restrict__ Bw,
                               float* __restrict__ C,
                               const float* __restrict__ bias,
                               int K, int lda, int ldb, int ldc) {
    const int lane = threadIdx.x & 31;
    const int r    = lane & 15;
    const int half = lane >> 4;
    const int n0   = blockIdx.x * (16 * NU);
    const int m0   = blockIdx.y * (16 * MU);

    const float* Ap[MU];
    const float* Bp[NU];
#pragma unroll
    for (int mi = 0; mi < MU; ++mi)
        Ap[mi] = A + (size_t)(m0 + 16 * mi + r) * lda + 2 * half;
#pragma unroll
    for (int ni = 0; ni < NU; ++ni)
        Bp[ni] = Bw + (size_t)(n0 + 16 * ni + r) * ldb + 2 * half;

    v8f acc[MU][NU];
#pragma unroll
    for (int mi = 0; mi < MU; ++mi)
#pragma unroll
        for (int ni = 0; ni < NU; ++ni)
            acc[mi][ni] = (v8f){};

    for (int k0 = 0; k0 < K; k0 += 16) {
        // Prefetch 64 floats (256B, 4 K-steps) ahead per stream (global_prefetch_b8:
        // speculative, no LOADcnt impact). Straight-line: no branches.
#pragma unroll
        for (int mi = 0; mi < MU; ++mi)
            __builtin_prefetch(Ap[mi] + k0 + 64, 0, 3);
#pragma unroll
        for (int ni = 0; ni < NU; ++ni)
            __builtin_prefetch(Bp[ni] + k0 + 64, 0, 3);

#pragma unroll
        for (int kk = 0; kk < 16; kk += 4) {
            const int k = k0 + kk;
            v2f a[MU], b[NU];
#pragma unroll
            for (int mi = 0; mi < MU; ++mi) a[mi] = *(const v2f*)(Ap[mi] + k);
#pragma unroll
            for (int ni = 0; ni < NU; ++ni) b[ni] = *(const v2f*)(Bp[ni] + k);
#pragma unroll
            for (int mi = 0; mi < MU; ++mi)
#pragma unroll
                for (int ni = 0; ni < NU; ++ni)
                    acc[mi][ni] = wmma_f32(a[mi], b[ni], acc[mi][ni]);
        }
    }

#pragma unroll
    for (int mi = 0; mi < MU; ++mi) {
#pragma unroll
        for (int ni = 0; ni < NU; ++ni) {
            const int col = n0 + 16 * ni + r;
            float bv = 0.f;
            if (EPI == 1) bv = bias[col];
#pragma unroll
            for (int i = 0; i < 8; ++i) {
                const int row = m0 + 16 * mi + i + 8 * half;
                float v = acc[mi][ni][i];
                if (EPI == 1) {
                    v += bv;
                    v = (v > 20.f) ? v : log1pf(expf(v));
                }
                C[(size_t)row * ldc + col] = v;
            }
        }
    }
}

// xc[b,l,d] = silu(conv_b[d] + sum_i xz[b, l-3+i, d] * conv_w[d][i])
// xz = first DINNER columns of the (2*DINNER)-wide in_proj output.
__global__ void conv_silu_kernel(const float* __restrict__ xzres,
                                 const float* __restrict__ conv_w,
                                 const float* __restrict__ conv_b,
                                 float* __restrict__ xc) {
    const int idx = blockIdx.x * blockDim.x + threadIdx.x;  // b*L*D + l*D + d
    const int d = idx & (DINNER - 1);
    const int l = (idx >> 11) & (L_DIM - 1);
    const int b = idx >> 21;

    float acc = conv_b[d];
#pragma unroll
    for (int i = 0; i < DCONV; ++i) {
        const int t = l - (DCONV - 1) + i;
        if (t >= 0)
            acc += xzres[(size_t)(b * L_DIM + t) * (2 * DINNER) + d] *
                   conv_w[d * DCONV + i];
    }
    xc[(size_t)idx] = acc / (1.f + expf(-acc));  // silu
}

// Selective scan + skip (xc*D) + silu(res) gate.
// Thread = one (b, d) channel; 16 states in registers; B_t/C_t staged in LDS.
// Writes gated y in place over the delta buffer.
__global__ void scan_kernel(const float* __restrict__ xc,
                            const float* __restrict__ xdbl,
                            float* __restrict__ delta_y,
                            const float* __restrict__ xzres,
                            const float* __restrict__ A_log,
                            const float* __restrict__ Dvec) {
    __shared__ float sB[DSTATE];
    __shared__ float sC[DSTATE];

    const int b = blockIdx.x >> 3;                       // 8 blocks per batch
    const int d = ((blockIdx.x & 7) << 8) + threadIdx.x; // 256 channels/block

    float Av[DSTATE], h[DSTATE];
#pragma unroll
    for (int n = 0; n < DSTATE; ++n) {
        Av[n] = -expf(A_log[d * DSTATE + n]);
        h[n]  = 0.f;
    }
    const float Dd = Dvec[d];

    for (int t = 0; t < L_DIM; ++t) {
        const size_t row = (size_t)(b * L_DIM + t);
        if (threadIdx.x < DSTATE)
            sB[threadIdx.x] = xdbl[row * 96 + DTRANK + threadIdx.x];
        else if (threadIdx.x < 2 * DSTATE)
            sC[threadIdx.x - DSTATE] =
                xdbl[row * 96 + DTRANK + DSTATE + (threadIdx.x - DSTATE)];
        __syncthreads();

        const float dlt = delta_y[row * DINNER + d];
        const float xv  = xc[row * DINNER + d];
        const float dx  = dlt * xv;
        float acc = 0.f;
#pragma unroll
        for (int n = 0; n < DSTATE; ++n) {
            h[n] = expf(dlt * Av[n]) * h[n] + dx * sB[n];
            acc += h[n] * sC[n];
        }
        const float rv = xzres[row * (2 * DINNER) + DINNER + d];
        const float yv = (acc + xv * Dd) * (rv / (1.f + expf(-rv)));

        __syncthreads();  // everyone done with sB/sC before next overwrite
        delta_y[row * DINNER + d] = yv;
    }
}

extern "C" void kernel_launch(void* const* d_in, const int* in_sizes, int n_in,
                              void* d_out, int out_size, void* d_ws, size_t ws_size,
                              hipStream_t stream) {
    (void)in_sizes; (void)n_in; (void)out_size; (void)ws_size;
    const float* x      = (const float*)d_in[0];
    const float* W_in   = (const float*)d_in[1];
    const float* conv_w = (const float*)d_in[2];
    const float* conv_b = (const float*)d_in[3];
    const float* W_x    = (const float*)d_in[4];
    const float* W_dt   = (const float*)d_in[5];
    const float* b_dt   = (const float*)d_in[6];
    const float* A_log  = (const float*)d_in[7];
    const float* Dv     = (const float*)d_in[8];
    const float* W_out  = (const float*)d_in[9];
    float* out = (float*)d_out;
    float* ws  = (float*)d_ws;

    float* xzres = ws;                                       // 2048 x 4096
    float* xc    = xzres + (size_t)M_ROWS * 2 * DINNER;      // 2048 x 2048
    float* xdbl  = xc    + (size_t)M_ROWS * DINNER;          // 2048 x 96
    float* delta = xdbl  + (size_t)M_ROWS * 96;              // 2048 x 2048 (then y)

    const dim3 wv(32);

    // 1) in_proj: (2048 x 4096) = x @ W_in^T   [32x64 tile/wave]
    gemm_nt_kernel<2, 4, 0><<<dim3(2 * DINNER / 64, M_ROWS / 32), wv, 0, stream>>>(
        x, W_in, xzres, nullptr, DMODEL, DMODEL, DMODEL, 2 * DINNER);

    // 2) causal depthwise conv + silu
    conv_silu_kernel<<<dim3(M_ROWS * DINNER / 256), dim3(256), 0, stream>>>(
        xzres, conv_w, conv_b, xc);

    // 3) x_proj: (2048 x 96) = xc @ W_x^T   [32x32 tile/wave, N=96=3*32]
    gemm_nt_kernel<2, 2, 0><<<dim3(96 / 32, M_ROWS / 32), wv, 0, stream>>>(
        xc, W_x, xdbl, nullptr, DINNER, DINNER, DINNER, 96);

    // 4) dt_proj + softplus: (2048 x 2048) = softplus(xdbl[:, :64] @ W_dt^T + b_dt)
    gemm_nt_kernel<2, 4, 1><<<dim3(DINNER / 64, M_ROWS / 32), wv, 0, stream>>>(
        xdbl, W_dt, delta, b_dt, DTRANK, 96, DTRANK, DINNER);

    // 5) selective scan + skip + gate (y overwrites delta buffer)
    scan_kernel<<<dim3(B_DIM * DINNER / 256), dim3(256), 0, stream>>>(
        xc, xdbl, delta, xzres, A_log, Dv);

    // 6) out_proj: (2048 x 1024) = y @ W_out^T
    gemm_nt_kernel<2, 4, 0><<<dim3(DMODEL / 64, M_ROWS / 32), wv, 0, stream>>>(
        delta, W_out, out, nullptr, DINNER, DINNER, DINNER, DMODEL);
}